// Model_6932077216072
// MI455X (gfx1250) — compile-verified
//
#include <hip/hip_runtime.h>
#include <math.h>

// ---------------- model dimensions ----------------
#define N_NODES 2048
#define T_TOK   4096
#define DB      768          // D_BERT
#define HID     256
#define GIN_D   832          // 768+32+32
#define H1      128          // per-direction GRU1 hidden
#define E_ENT   256
#define P_PAIR  2048
#define K_PATH  3
#define LP_LEN  16
#define R_REL   97
#define SEQ2    (P_PAIR*K_PATH)   // 6144
#define ATT_D   1056

typedef __attribute__((ext_vector_type(2))) float v2f;
typedef __attribute__((ext_vector_type(8))) float v8f;

__device__ __forceinline__ float sigmoidf_(float x) { return 1.f / (1.f + __expf(-x)); }

// =====================================================================
// Generic fp32 WMMA GEMM: C = act(((A @ B(^T)) + bias) * rowscale)
//  BT=1: B is a (Nc x Kc) weight matrix, compute A @ B^T
//  GATHER=1: A row r is A[gidx[r*gstride+goff]]
// One wave computes a 32x32 C tile as 2x2 WMMA 16x16 sub-tiles
// (V_WMMA_F32_16X16X4_F32): 4 independent accumulation chains pipeline
// the matrix unit, and A/B loads are shared across sub-tiles
// (8 FLOP/byte from L2 vs 4 for a single 16x16 tile).
// Operand layout per CDNA5 ISA 7.12.2:
//   A 16x4 : lanes 0-15 M=lane, Kpair {0,1}; lanes 16-31 M=lane-16, Kpair {2,3}
//   B 4x16 : mirrored (N=lane%16, Kpair by lane/16)
//   C 16x16: VGPR j -> row j (lanes 0-15) / row j+8 (lanes 16-31), col = lane%16
// =====================================================================
template<int BT, int GATHER>
__global__ void wmma_gemm(const float* __restrict__ A, const float* __restrict__ B,
                          const float* __restrict__ bias, const float* __restrict__ rowscale,
                          float* __restrict__ C, int M, int Nc, int Kc,
                          int lda, int ldb, int ldc, int act,
                          const int* __restrict__ gidx, int gstride, int goff)
{
    const int tilesN = Nc >> 5;                 // 32-wide tiles
    const int tiles  = (M >> 5) * tilesN;
    const int tile   = blockIdx.x * 4 + (threadIdx.x >> 5);
    if (tile >= tiles) return;
    const int tm   = (tile / tilesN) << 5;
    const int tn   = (tile % tilesN) << 5;
    const int lane = threadIdx.x & 31;
    const int l16  = lane & 15;
    const int half = lane >> 4;

    int ar0 = tm + l16;
    int ar1 = tm + 16 + l16;
    if (GATHER) {
        ar0 = gidx[(size_t)ar0 * gstride + goff];
        ar1 = gidx[(size_t)ar1 * gstride + goff];
    }
    const float* Ap0 = A + (size_t)ar0 * lda + 2 * half;
    const float* Ap1 = A + (size_t)ar1 * lda + 2 * half;

    v8f acc[2][2];
    #pragma unroll
    for (int mi = 0; mi < 2; ++mi)
        #pragma unroll
        for (int ni = 0; ni < 2; ++ni)
            #pragma unroll
            for (int q = 0; q < 8; ++q) acc[mi][ni][q] = 0.f;

    if (BT) {
        const float* Bp0 = B + (size_t)(tn + l16) * ldb + 2 * half;
        const float* Bp1 = Bp0 + (size_t)16 * ldb;
        #pragma unroll 4
        for (int k = 0; k < Kc; k += 4) {
            v2f a0; a0.x = Ap0[k]; a0.y = Ap0[k + 1];
            v2f a1; a1.x = Ap1[k]; a1.y = Ap1[k + 1];
            v2f b0; b0.x = Bp0[k]; b0.y = Bp0[k + 1];
            v2f b1; b1.x = Bp1[k]; b1.y = Bp1[k + 1];
            acc[0][0] = __builtin_amdgcn_wmma_f32_16x16x4_f32(false, a0, false, b0, (short)0, acc[0][0], false, false);
            acc[0][1] = __builtin_amdgcn_wmma_f32_16x16x4_f32(false, a0, false, b1, (short)0, acc[0][1], false, false);
            acc[1][0] = __builtin_amdgcn_wmma_f32_16x16x4_f32(false, a1, false, b0, (short)0, acc[1][0], false, false);
            acc[1][1] = __builtin_amdgcn_wmma_f32_16x16x4_f32(false, a1, false, b1, (short)0, acc[1][1], false, false);
        }
    } else {
        #pragma unroll 4
        for (int k = 0; k < Kc; k += 4) {
            v2f a0; a0.x = Ap0[k]; a0.y = Ap0[k + 1];
            v2f a1; a1.x = Ap1[k]; a1.y = Ap1[k + 1];
            const int kr = k + 2 * half;
            const float* Br0 = B + (size_t)kr * ldb + tn + l16;
            const float* Br1 = B + (size_t)(kr + 1) * ldb + tn + l16;
            v2f b0; b0.x = Br0[0];  b0.y = Br1[0];
            v2f b1; b1.x = Br0[16]; b1.y = Br1[16];
            acc[0][0] = __builtin_amdgcn_wmma_f32_16x16x4_f32(false, a0, false, b0, (short)0, acc[0][0], false, false);
            acc[0][1] = __builtin_amdgcn_wmma_f32_16x16x4_f32(false, a0, false, b1, (short)0, acc[0][1], false, false);
            acc[1][0] = __builtin_amdgcn_wmma_f32_16x16x4_f32(false, a1, false, b0, (short)0, acc[1][0], false, false);
            acc[1][1] = __builtin_amdgcn_wmma_f32_16x16x4_f32(false, a1, false, b1, (short)0, acc[1][1], false, false);
        }
    }

    #pragma unroll
    for (int mi = 0; mi < 2; ++mi) {
        #pragma unroll
        for (int ni = 0; ni < 2; ++ni) {
            #pragma unroll
            for (int j = 0; j < 8; ++j) {
                const int row = tm + mi * 16 + j + half * 8;
                const int col = tn + ni * 16 + l16;
                float v = acc[mi][ni][j];
                if (bias)     v += bias[col];
                if (rowscale) v *= rowscale[row];
                if (act == 1)      v = v > 0.f ? v : 0.f;
                else if (act == 2) v = sigmoidf_(v);
                C[(size_t)row * ldc + col] = v;
            }
        }
    }
}

// ---------------- span mean + type/common embedding concat -> x (N x 832) ----
__global__ void build_x(const float* __restrict__ tokens, const int* __restrict__ start,
                        const int* __restrict__ len, const int* __restrict__ types,
                        const int* __restrict__ commons, const float* __restrict__ type_emb,
                        const float* __restrict__ com_emb, float* __restrict__ x)
{
    int idx = blockIdx.x * blockDim.x + threadIdx.x;
    if (idx >= N_NODES * GIN_D) return;
    int n = idx / GIN_D, d = idx % GIN_D;
    float v;
    if (d < DB) {
        int s = start[n];
        int e = s + len[n]; if (e > T_TOK - 1) e = T_TOK - 1;
        float sum = 0.f;
        for (int t = s; t <= e; ++t) sum += tokens[(size_t)t * DB + d];
        v = sum / (float)(e - s + 1);
    } else if (d < DB + 32) {
        v = type_emb[types[n] * 32 + (d - DB)];
    } else {
        v = com_emb[commons[n] * 32 + (d - DB - 32)];
    }
    x[idx] = v;
}

// ---------------- 1 / (adj row sum + 1) ----------------
__global__ void row_inv_denom(const float* __restrict__ adj, float* __restrict__ inv)
{
    int n = blockIdx.x * blockDim.x + threadIdx.x;
    if (n >= N_NODES) return;
    float s = 0.f;
    const float* r = adj + (size_t)n * N_NODES;
    for (int c = 0; c < N_NODES; ++c) s += r[c];
    inv[n] = 1.f / (s + 1.f);
}

// ---------------- GRU1: serial bidirectional recurrence (gi precomputed) ----
__global__ void gru1_seq(const float* __restrict__ gi_f, const float* __restrict__ gi_b,
                         const float* __restrict__ Whh_f, const float* __restrict__ bhh_f,
                         const float* __restrict__ Whh_b, const float* __restrict__ bhh_b,
                         float* __restrict__ out /* N x 256 */)
{
    const int dir = blockIdx.x;
    const float* gi  = dir ? gi_b  : gi_f;
    const float* Whh = dir ? Whh_b : Whh_f;
    const float* bhh = dir ? bhh_b : bhh_f;
    __shared__ float h[H1];
    __shared__ float gh[3 * H1];
    const int j = threadIdx.x;  // 0..383
    if (j < H1) h[j] = 0.f;
    __syncthreads();
    for (int step = 0; step < N_NODES; ++step) {
        const int t = dir ? (N_NODES - 1 - step) : step;
        float s = bhh[j];
        const float* wr = Whh + (size_t)j * H1;
        #pragma unroll 4
        for (int c = 0; c < H1; ++c) s += wr[c] * h[c];
        gh[j] = s;
        __syncthreads();
        if (j < H1) {
            const float* g = gi + (size_t)t * (3 * H1);
            float r = sigmoidf_(g[j] + gh[j]);
            float z = sigmoidf_(g[H1 + j] + gh[H1 + j]);
            float n = tanhf(g[2 * H1 + j] + r * gh[2 * H1 + j]);
            float hn = (1.f - z) * n + z * h[j];
            h[j] = hn;
            out[(size_t)t * HID + dir * H1 + j] = hn;
        }
        __syncthreads();
    }
}

// ---------------- GCN helpers ----------------
__global__ void build_cat1(const float* __restrict__ xin, const float* __restrict__ g0,
                           float* __restrict__ cat1)
{
    int idx = blockIdx.x * blockDim.x + threadIdx.x;
    if (idx >= N_NODES * 384) return;
    int n = idx / 384, c = idx % 384;
    cat1[idx] = (c < HID) ? xin[(size_t)n * HID + c] : g0[(size_t)n * 128 + (c - HID)];
}
__global__ void gcn_go(const float* __restrict__ g0, const float* __restrict__ g1,
                       const float* __restrict__ xin, float* __restrict__ go)
{
    int idx = blockIdx.x * blockDim.x + threadIdx.x;
    if (idx >= N_NODES * HID) return;
    int n = idx / HID, c = idx % HID;
    float v = (c < 128) ? g0[(size_t)n * 128 + c] : g1[(size_t)n * 128 + (c - 128)];
    go[idx] = v + xin[idx];
}
__global__ void gcn_out(const float* __restrict__ gate, const float* __restrict__ wo,
                        const float* __restrict__ go, float* __restrict__ xout)
{
    int idx = blockIdx.x * blockDim.x + threadIdx.x;
    if (idx >= N_NODES * HID) return;
    float g = gate[idx];
    xout[idx] = g * wo[idx] + (1.f - g) * go[idx];
}

// ---------------- entity mean over 4 mentions x 4 tokens ----------------
__global__ void entity_mean(const float* __restrict__ words, const int* __restrict__ mention_pos,
                            float* __restrict__ ent)
{
    int idx = blockIdx.x * blockDim.x + threadIdx.x;
    if (idx >= E_ENT * HID) return;
    int e = idx / HID, d = idx % HID;
    float s = 0.f;
    for (int m = 0; m < 16; ++m)
        s += words[(size_t)mention_pos[e * 16 + m] * HID + d];
    ent[idx] = s * (1.f / 16.f);
}

// ---------------- zero fill ----------------
__global__ void fill_zero(float* __restrict__ p, int n)
{
    int i = blockIdx.x * blockDim.x + threadIdx.x;
    if (i < n) p[i] = 0.f;
}

// ---------------- GRU2 gate update: h = (1-z)*n + z*h ----------------
__global__ void gru2_update(const float* __restrict__ gi, const float* __restrict__ gh,
                            float* __restrict__ h)
{
    int idx = blockIdx.x * blockDim.x + threadIdx.x;
    if (idx >= SEQ2 * HID) return;
    int s = idx / HID, j = idx % HID;
    const float* gis = gi + (size_t)s * 768;
    const float* ghs = gh + (size_t)s * 768;
    float r = sigmoidf_(gis[j] + ghs[j]);
    float z = sigmoidf_(gis[HID + j] + ghs[HID + j]);
    float n = tanhf(gis[2 * HID + j] + r * ghs[2 * HID + j]);
    h[idx] = (1.f - z) * n + z * h[idx];
}

// ---------------- attention over K=3 paths, build head/tail (P x 512) ------
__global__ void attention_kernel(const float* __restrict__ ent, const int* __restrict__ head_idx,
                                 const int* __restrict__ tail_idx, const float* __restrict__ head_path,
                                 const float* __restrict__ tail_path, const float* __restrict__ dis_emb,
                                 const int* __restrict__ distances, const float* __restrict__ w_att,
                                 const float* __restrict__ b_att, float* __restrict__ head,
                                 float* __restrict__ tail)
{
    const int p = blockIdx.x;
    const int tid = threadIdx.x;  // 256 threads
    __shared__ float red[256];
    __shared__ float sc[4];
    const int hi = head_idx[p], ti = tail_idx[p];
    const float* he = ent + (size_t)hi * HID;
    const float* te = ent + (size_t)ti * HID;
    for (int k = 0; k < K_PATH; ++k) {
        const float* hp = head_path + (size_t)(p * K_PATH + k) * HID;
        const float* tp = tail_path + (size_t)(p * K_PATH + k) * HID;
        const float* de = dis_emb + (size_t)distances[p * K_PATH + k] * 32;
        float s = 0.f;
        for (int d = tid; d < ATT_D; d += 256) {
            float f = (d < 256) ? he[d] : (d < 512) ? te[d - 256] : (d < 768) ? hp[d - 512]
                      : (d < 1024) ? tp[d - 768] : de[d - 1024];
            s += w_att[d] * f;
        }
        red[tid] = s; __syncthreads();
        for (int st = 128; st > 0; st >>= 1) {
            if (tid < st) red[tid] += red[tid + st];
            __syncthreads();
        }
        if (tid == 0) sc[k] = red[0] + b_att[0];
        __syncthreads();
    }
    if (tid == 0) {
        float m  = fmaxf(sc[0], fmaxf(sc[1], sc[2]));
        float e0 = __expf(sc[0] - m), e1 = __expf(sc[1] - m), e2 = __expf(sc[2] - m);
        float inv = 1.f / (e0 + e1 + e2);
        sc[0] = e0 * inv; sc[1] = e1 * inv; sc[2] = e2 * inv;
    }
    __syncthreads();
    if (tid < HID) {
        float hemb = 0.f, temb = 0.f;
        for (int k = 0; k < K_PATH; ++k) {
            hemb += sc[k] * head_path[(size_t)(p * K_PATH + k) * HID + tid];
            temb += sc[k] * tail_path[(size_t)(p * K_PATH + k) * HID + tid];
        }
        head[(size_t)p * 512 + tid]       = he[tid];
        head[(size_t)p * 512 + 256 + tid] = hemb;
        tail[(size_t)p * 512 + tid]       = te[tid];
        tail[(size_t)p * 512 + 256 + tid] = temb;
    }
}

// ---------------- fused bilinear: logits[p,r] = head[p] . (Wb[r] @ tail[p]) + bb[r]
// One wave per (r, 16-row p tile). C[p,i] = sum_j tail[p,j]*Wb[r,i,j] via WMMA,
// reduced against head in-register (no (P,R,512) intermediate). Two i-chunks
// per pass share the A (tail) loads -> two concurrent WMMA chains.
__global__ void bilinear_kernel(const float* __restrict__ head, const float* __restrict__ tail,
                                const float* __restrict__ Wb, const float* __restrict__ bb,
                                float* __restrict__ out)
{
    const int r    = blockIdx.x;
    const int pt   = blockIdx.y * 4 + (threadIdx.x >> 5);
    const int row0 = pt << 4;
    const int lane = threadIdx.x & 31;
    const int l16  = lane & 15;
    const int half = lane >> 4;
    const float* W = Wb + (size_t)r * 512 * 512;

    float acc2[8];
    #pragma unroll
    for (int j = 0; j < 8; ++j) acc2[j] = 0.f;

    const float* Ap = tail + (size_t)(row0 + l16) * 512 + 2 * half;
    for (int i0 = 0; i0 < 512; i0 += 32) {
        v8f c0, c1;
        #pragma unroll
        for (int q = 0; q < 8; ++q) { c0[q] = 0.f; c1[q] = 0.f; }
        const float* Bp0 = W + (size_t)(i0 + l16) * 512 + 2 * half;
        const float* Bp1 = Bp0 + (size_t)16 * 512;
        #pragma unroll 4
        for (int k = 0; k < 512; k += 4) {
            v2f a;  a.x  = Ap[k];  a.y  = Ap[k + 1];
            v2f b0; b0.x = Bp0[k]; b0.y = Bp0[k + 1];
            v2f b1; b1.x = Bp1[k]; b1.y = Bp1[k + 1];
            c0 = __builtin_amdgcn_wmma_f32_16x16x4_f32(false, a, false, b0, (short)0, c0, false, false);
            c1 = __builtin_amdgcn_wmma_f32_16x16x4_f32(false, a, false, b1, (short)0, c1, false, false);
        }
        #pragma unroll
        for (int j = 0; j < 8; ++j) {
            const int prow = row0 + j + half * 8;
            acc2[j] += c0[j] * head[(size_t)prow * 512 + i0 + l16]
                     + c1[j] * head[(size_t)prow * 512 + i0 + 16 + l16];
        }
    }
    #pragma unroll
    for (int j = 0; j < 8; ++j) {
        float v = acc2[j];
        for (int off = 8; off > 0; off >>= 1) v += __shfl_xor(v, off, 16);
        if (l16 == 0)
            out[(size_t)(row0 + j + half * 8) * R_REL + r] = v + bb[r];
    }
}

// =====================================================================
// Host side
// =====================================================================
static inline void g_nt(const float* A, int lda, const float* B, int ldb, const float* bias,
                        const float* rs, float* C, int ldc, int M, int Nc, int Kc, int act,
                        hipStream_t s)
{
    int tiles = (M / 32) * (Nc / 32);
    wmma_gemm<1, 0><<<(tiles + 3) / 4, 128, 0, s>>>(A, B, bias, rs, C, M, Nc, Kc, lda, ldb, ldc,
                                                    act, nullptr, 0, 0);
}
static inline void g_nn(const float* A, int lda, const float* B, int ldb, const float* bias,
                        const float* rs, float* C, int ldc, int M, int Nc, int Kc, int act,
                        hipStream_t s)
{
    int tiles = (M / 32) * (Nc / 32);
    wmma_gemm<0, 0><<<(tiles + 3) / 4, 128, 0, s>>>(A, B, bias, rs, C, M, Nc, Kc, lda, ldb, ldc,
                                                    act, nullptr, 0, 0);
}
static inline void g_nt_gather(const float* A, int lda, const float* B, int ldb, const float* bias,
                               float* C, int ldc, int M, int Nc, int Kc, const int* gidx,
                               int gstride, int goff, hipStream_t s)
{
    int tiles = (M / 32) * (Nc / 32);
    wmma_gemm<1, 1><<<(tiles + 3) / 4, 128, 0, s>>>(A, B, bias, nullptr, C, M, Nc, Kc, lda, ldb,
                                                    ldc, 0, gidx, gstride, goff);
}

// leaf sizes (elements) in flattened setup_inputs order
static const long long LS[50] = {
    (long long)T_TOK * DB, N_NODES, N_NODES, N_NODES, N_NODES,
    (long long)N_NODES * N_NODES, E_ENT * 16, P_PAIR, P_PAIR,
    (long long)P_PAIR * K_PATH * LP_LEN, P_PAIR * K_PATH,
    320, 320, 640,
    384LL * 832, 384 * 128, 384, 384, 384LL * 832, 384 * 128, 384, 384,
    768LL * 256, 768LL * 256, 768, 768, 768LL * 256, 768LL * 256, 768, 768,
    128LL * 256, 128, 128LL * 384, 128, 256LL * 256, 256, 256LL * 256, 256,
    128LL * 256, 128, 128LL * 384, 128, 256LL * 256, 256, 256LL * 256, 256,
    1056, 1,
    (long long)R_REL * 512 * 512, R_REL};
static const int GSTART[18] = {0,1,2,3,4,5,6,7,8,9,10,11,14,22,30,38,46,48};
static const int GLEN[18]   = {1,1,1,1,1,1,1,1,1,1,1, 3, 8, 8, 8, 8, 2, 2};

static void run_gcn(const void* const* L, int base, const float* adj, const float* inv_denom,
                    const float* xin, float* xout, float* axbuf, float* g0, float* cat1, float* g1,
                    float* go, float* gate, float* wo, hipStream_t s)
{
    const float* W0 = (const float*)L[base + 0]; const float* b0 = (const float*)L[base + 1];
    const float* W1 = (const float*)L[base + 2]; const float* b1 = (const float*)L[base + 3];
    const float* Wg = (const float*)L[base + 4]; const float* bg = (const float*)L[base + 5];
    const float* Wo = (const float*)L[base + 6]; const float* bo = (const float*)L[base + 7];
    g_nn(adj, N_NODES, xin, HID, nullptr, nullptr, axbuf, HID, N_NODES, HID, N_NODES, 0, s);
    g_nt(axbuf, HID, W0, HID, b0, inv_denom, g0, 128, N_NODES, 128, HID, 1, s);
    build_cat1<<<(N_NODES * 384 + 255) / 256, 256, 0, s>>>(xin, g0, cat1);
    g_nn(adj, N_NODES, cat1, 384, nullptr, nullptr, axbuf, 384, N_NODES, 384, N_NODES, 0, s);
    g_nt(axbuf, 384, W1, 384, b1, inv_denom, g1, 128, N_NODES, 128, 384, 1, s);
    gcn_go<<<(N_NODES * HID + 255) / 256, 256, 0, s>>>(g0, g1, xin, go);
    g_nt(go, HID, Wg, HID, bg, nullptr, gate, HID, N_NODES, HID, HID, 2, s);
    g_nt(go, HID, Wo, HID, bo, nullptr, wo, HID, N_NODES, HID, HID, 0, s);
    gcn_out<<<(N_NODES * HID + 255) / 256, 256, 0, s>>>(gate, wo, go, xout);
}

extern "C" void kernel_launch(void* const* d_in, const int* in_sizes, int n_in,
                              void* d_out, int out_size, void* d_ws, size_t ws_size,
                              hipStream_t stream)
{
    (void)in_sizes; (void)out_size; (void)ws_size;
    // -------- resolve leaves (handles flattened-50 or dict-18 layouts) -----
    const void* L[50];
    if (n_in >= 50) {
        for (int i = 0; i < 50; ++i) L[i] = d_in[i];
    } else {
        for (int g = 0; g < 18 && g < n_in; ++g) {
            const char* base = (const char*)d_in[g];
            long long off = 0;
            for (int k = 0; k < GLEN[g]; ++k) {
                int li = GSTART[g] + k;
                L[li] = base + off * 4;
                off += LS[li];
            }
        }
    }
    const float* tokens   = (const float*)L[0];
    const int*   sp_start = (const int*)L[1];
    const int*   sp_len   = (const int*)L[2];
    const int*   types    = (const int*)L[3];
    const int*   commons  = (const int*)L[4];
    const float* adj      = (const float*)L[5];
    const int*   mpos     = (const int*)L[6];
    const int*   head_idx = (const int*)L[7];
    const int*   tail_idx = (const int*)L[8];
    const int*   paths    = (const int*)L[9];
    const int*   dists    = (const int*)L[10];
    const float* type_emb = (const float*)L[11];
    const float* com_emb  = (const float*)L[12];
    const float* dis_emb  = (const float*)L[13];

    // -------- workspace bump allocator (fp32 units, 256B aligned) ----------
    float* w = (float*)d_ws;
    size_t o = 0;
    auto alloc = [&](size_t n) { float* p = w + o; o += (n + 63) & ~(size_t)63; return p; };
    float* x       = alloc((size_t)N_NODES * GIN_D);
    float* gi1f    = alloc((size_t)N_NODES * 384);
    float* gi1b    = alloc((size_t)N_NODES * 384);
    float* wordsA  = alloc((size_t)N_NODES * HID);
    float* wordsB  = alloc((size_t)N_NODES * HID);
    float* invden  = alloc(N_NODES);
    float* axbuf   = alloc((size_t)N_NODES * 384);
    float* g0buf   = alloc((size_t)N_NODES * 128);
    float* cat1    = alloc((size_t)N_NODES * 384);
    float* g1buf   = alloc((size_t)N_NODES * 128);
    float* gobuf   = alloc((size_t)N_NODES * HID);
    float* gatebuf = alloc((size_t)N_NODES * HID);
    float* wobuf   = alloc((size_t)N_NODES * HID);
    float* ent     = alloc((size_t)E_ENT * HID);
    float* hF      = alloc((size_t)SEQ2 * HID);   // forward final -> tail_path
    float* hB      = alloc((size_t)SEQ2 * HID);   // backward final -> head_path
    float* gi2     = alloc((size_t)SEQ2 * 768);
    float* gh2     = alloc((size_t)SEQ2 * 768);
    float* headb   = alloc((size_t)P_PAIR * 512);
    float* tailb   = alloc((size_t)P_PAIR * 512);

    // -------- 1. word/type/common features ---------------------------------
    build_x<<<(N_NODES * GIN_D + 255) / 256, 256, 0, stream>>>(
        tokens, sp_start, sp_len, types, commons, type_emb, com_emb, x);
    row_inv_denom<<<(N_NODES + 255) / 256, 256, 0, stream>>>(adj, invden);

    // -------- 2. GRU1: batched input projection + serial recurrence --------
    g_nt(x, GIN_D, (const float*)L[14], GIN_D, (const float*)L[16], nullptr, gi1f, 384,
         N_NODES, 384, GIN_D, 0, stream);
    g_nt(x, GIN_D, (const float*)L[18], GIN_D, (const float*)L[20], nullptr, gi1b, 384,
         N_NODES, 384, GIN_D, 0, stream);
    gru1_seq<<<2, 384, 0, stream>>>(gi1f, gi1b, (const float*)L[15], (const float*)L[17],
                                    (const float*)L[19], (const float*)L[21], wordsA);

    // -------- 3. two GCN blocks -------------------------------------------
    run_gcn(L, 30, adj, invden, wordsA, wordsB, axbuf, g0buf, cat1, g1buf, gobuf, gatebuf, wobuf, stream);
    run_gcn(L, 38, adj, invden, wordsB, wordsA, axbuf, g0buf, cat1, g1buf, gobuf, gatebuf, wobuf, stream);
    const float* words = wordsA;

    // -------- 4. entity pooling -------------------------------------------
    entity_mean<<<(E_ENT * HID + 255) / 256, 256, 0, stream>>>(words, mpos, ent);

    // -------- 5. GRU2 over 6144 paths of length 16, both directions --------
    fill_zero<<<(SEQ2 * HID + 255) / 256, 256, 0, stream>>>(hF, SEQ2 * HID);
    fill_zero<<<(SEQ2 * HID + 255) / 256, 256, 0, stream>>>(hB, SEQ2 * HID);
    const float* Wih2f = (const float*)L[22]; const float* Whh2f = (const float*)L[23];
    const float* bih2f = (const float*)L[24]; const float* bhh2f = (const float*)L[25];
    const float* Wih2b = (const float*)L[26]; const float* Whh2b = (const float*)L[27];
    const float* bih2b = (const float*)L[28]; const float* bhh2b = (const float*)L[29];
    for (int t = 0; t < LP_LEN; ++t) {
        g_nt_gather(words, HID, Wih2f, HID, bih2f, gi2, 768, SEQ2, 768, HID, paths, LP_LEN, t, stream);
        g_nt(hF, HID, Whh2f, HID, bhh2f, nullptr, gh2, 768, SEQ2, 768, HID, 0, stream);
        gru2_update<<<(SEQ2 * HID + 255) / 256, 256, 0, stream>>>(gi2, gh2, hF);
    }
    for (int t = 0; t < LP_LEN; ++t) {
        g_nt_gather(words, HID, Wih2b, HID, bih2b, gi2, 768, SEQ2, 768, HID, paths, LP_LEN,
                    LP_LEN - 1 - t, stream);
        g_nt(hB, HID, Whh2b, HID, bhh2b, nullptr, gh2, 768, SEQ2, 768, HID, 0, stream);
        gru2_update<<<(SEQ2 * HID + 255) / 256, 256, 0, stream>>>(gi2, gh2, hB);
    }

    // -------- 6. attention over K paths, build head/tail -------------------
    attention_kernel<<<P_PAIR, 256, 0, stream>>>(ent, head_idx, tail_idx, hB /*head_path*/,
                                                 hF /*tail_path*/, dis_emb, dists,
                                                 (const float*)L[46], (const float*)L[47],
                                                 headb, tailb);

    // -------- 7. fused bilinear scorer -> logits (P x 97) ------------------
    bilinear_kernel<<<dim3(R_REL, P_PAIR / 64), 128, 0, stream>>>(
        headb, tailb, (const float*)L[48], (const float*)L[49], (float*)d_out);
}